// Model2_HSinner_7756710937248
// MI455X (gfx1250) — compile-verified
//
#include <hip/hip_runtime.h>
#include <hip/hip_bf16.h>

typedef __attribute__((ext_vector_type(2))) float v2f;
typedef __attribute__((ext_vector_type(8))) float v8f;

// ---------------------------------------------------------------------------
// Kernel 1: batched MLP via V_WMMA_F32_16X16X4_F32.
// grid = 16 blocks (8 row-tiles x 2 inputs), block = 32 threads (one wave).
// Weights are staged zero-padded into LDS once, so all WMMA fragment loads in
// the inner loops are unconditional ds_loads (no exec-mask branching).
// ---------------------------------------------------------------------------
__global__ __launch_bounds__(32) void mlp_wmma_kernel(
    const float* __restrict__ x1, const float* __restrict__ x2,
    const float* __restrict__ W1, const float* __restrict__ b1,
    const float* __restrict__ W2, const float* __restrict__ b2,
    const float* __restrict__ W3, const float* __restrict__ b3,
    float* __restrict__ angles)
{
    const int bx   = blockIdx.x;
    const int tile = bx & 7;      // which 16-row tile of B=128
    const int src  = bx >> 3;     // 0 -> x1, 1 -> x2
    const float* X = src ? x2 : x1;

    const int lane = threadIdx.x;      // 0..31
    const int half = lane >> 4;        // 0: K pair {0,1}, 1: K pair {2,3}
    const int l16  = lane & 15;

    __shared__ float W1p[8][32];       // zero padded N: 20 -> 32
    __shared__ float W2p[20][32];
    __shared__ float W3p[20][16];
    __shared__ float b1s[32], b2s[32], b3s[16];
    __shared__ float h[2][16][32];     // ping-pong hidden tiles

    // ---- stage weights (guards only here, once) ----
    for (int idx = lane; idx < 8 * 32; idx += 32) {
        const int k = idx >> 5, n = idx & 31;
        W1p[k][n] = (n < 20) ? W1[k * 20 + n] : 0.0f;
    }
    for (int idx = lane; idx < 20 * 32; idx += 32) {
        const int k = idx >> 5, n = idx & 31;
        W2p[k][n] = (n < 20) ? W2[k * 20 + n] : 0.0f;
    }
    for (int idx = lane; idx < 20 * 16; idx += 32) {
        W3p[idx >> 4][idx & 15] = W3[idx];
    }
    b1s[lane] = (lane < 20) ? b1[lane] : 0.0f;
    b2s[lane] = (lane < 20) ? b2[lane] : 0.0f;
    if (lane < 16) b3s[lane] = b3[lane];
    __syncthreads();

    // ---------------- Layer 1: (16x8) @ (8x20) + b1, relu ----------------
    #pragma unroll
    for (int n0 = 0; n0 < 32; n0 += 16) {
        v8f c = {};
        const int n = n0 + l16;
        #pragma unroll
        for (int k0 = 0; k0 < 8; k0 += 4) {
            const int ka = k0 + half * 2;
            v2f a, bf;
            a.x  = X[(tile * 16 + l16) * 8 + ka];
            a.y  = X[(tile * 16 + l16) * 8 + ka + 1];
            bf.x = W1p[ka][n];
            bf.y = W1p[ka + 1][n];
            c = __builtin_amdgcn_wmma_f32_16x16x4_f32(false, a, false, bf,
                                                      (short)0, c, false, false);
        }
        const float bias = b1s[n];
        #pragma unroll
        for (int r = 0; r < 8; ++r) {
            float v = c[r] + bias;
            h[0][r + half * 8][n] = v > 0.0f ? v : 0.0f;
        }
    }
    __syncthreads();

    // ---------------- Layer 2: (16x20) @ (20x20) + b2, relu --------------
    #pragma unroll
    for (int n0 = 0; n0 < 32; n0 += 16) {
        v8f c = {};
        const int n = n0 + l16;
        #pragma unroll
        for (int k0 = 0; k0 < 20; k0 += 4) {
            const int ka = k0 + half * 2;
            v2f a, bf;
            a.x  = h[0][l16][ka];
            a.y  = h[0][l16][ka + 1];
            bf.x = W2p[ka][n];
            bf.y = W2p[ka + 1][n];
            c = __builtin_amdgcn_wmma_f32_16x16x4_f32(false, a, false, bf,
                                                      (short)0, c, false, false);
        }
        const float bias = b2s[n];
        #pragma unroll
        for (int r = 0; r < 8; ++r) {
            float v = c[r] + bias;
            h[1][r + half * 8][n] = v > 0.0f ? v : 0.0f;
        }
    }
    __syncthreads();

    // ---------------- Layer 3: (16x20) @ (20x16) + b3 --------------------
    {
        v8f c = {};
        const int n = l16;  // N = 16 exactly
        #pragma unroll
        for (int k0 = 0; k0 < 20; k0 += 4) {
            const int ka = k0 + half * 2;
            v2f a, bf;
            a.x  = h[1][l16][ka];
            a.y  = h[1][l16][ka + 1];
            bf.x = W3p[ka][n];
            bf.y = W3p[ka + 1][n];
            c = __builtin_amdgcn_wmma_f32_16x16x4_f32(false, a, false, bf,
                                                      (short)0, c, false, false);
        }
        const float bias = b3s[n];
        #pragma unroll
        for (int r = 0; r < 8; ++r) {
            const int m = r + half * 8;
            angles[src * 2048 + (tile * 16 + m) * 16 + n] = c[r] + bias;
        }
    }
}

// ---------------------------------------------------------------------------
// Kernel 2: per-batch circuit simulation entirely in LDS.
// grid = 128 (one workgroup per batch element), block = 256 threads.
// 16 columns x 2 circuits per chunk (32 KB LDS), 16 chunks.
// The 8 ring-CNOTs collapse into ONE fixed index permutation (prefix-XOR in
// wire space): 1 read pass + barrier + 1 permuted write pass, instead of
// 8 barriered swap passes. RY steps for both circuits share one barrier.
// ---------------------------------------------------------------------------
__global__ __launch_bounds__(256) void circuit_kernel(
    const float* __restrict__ angles, float* __restrict__ out)
{
    const int b   = blockIdx.x;
    const int tid = threadIdx.x;

    __shared__ float st[2][16][256];   // [circuit][col][row]  (32 KB)
    __shared__ float red[256];
    __shared__ float cth[2][16], sth[2][16];
    __shared__ int   perm[256];        // composed ring-CNOT permutation

    // cos/sin of the 32 half-angles (once; LDS so no register spills)
    if (tid < 32) {
        const int s = tid >> 4, g = tid & 15;
        const float ha = angles[s * 2048 + b * 16 + g] * 0.5f;
        __sincosf(ha, &sth[s][g], &cth[s][g]);
    }
    // perm[r] = index after CNOT(0,1),CNOT(1,2),...,CNOT(7,0) applied to r
    {
        int r = tid;
        #pragma unroll
        for (int w = 0; w < 8; ++w) {
            const int bc = 7 - w;
            const int bt = 7 - ((w + 1) & 7);
            if ((r >> bc) & 1) r ^= (1 << bt);
        }
        perm[tid] = r;
    }
    __syncthreads();

    float acc = 0.0f;

    for (int chunk = 0; chunk < 16; ++chunk) {
        // init basis columns chunk*16 .. chunk*16+15 for both circuits
        for (int e = tid; e < 2 * 16 * 256; e += 256) {
            const int s   = e >> 12;
            const int col = (e >> 8) & 15;
            const int row = e & 255;
            st[s][col][row] = (row == chunk * 16 + col) ? 1.0f : 0.0f;
        }
        __syncthreads();

        for (int layer = 0; layer < 2; ++layer) {
            // ---- RY on wires 0..7 (wire w <-> row bit 7-w); both circuits
            //      between barriers (disjoint LDS regions) ----
            for (int w = 0; w < 8; ++w) {
                const int bit  = 7 - w;
                const int mlow = (1 << bit) - 1;
                #pragma unroll
                for (int s = 0; s < 2; ++s) {
                    const float cg = cth[s][layer * 8 + w];
                    const float sg = sth[s][layer * 8 + w];
                    for (int p = tid; p < 16 * 128; p += 256) {
                        const int col = p >> 7;
                        const int pr  = p & 127;
                        const int low = pr & mlow;
                        const int i0  = ((pr - low) << 1) | low;
                        const int i1  = i0 | (1 << bit);
                        const float v0 = st[s][col][i0];
                        const float v1 = st[s][col][i1];
                        st[s][col][i0] = cg * v0 - sg * v1;
                        st[s][col][i1] = sg * v0 + cg * v1;
                    }
                }
                __syncthreads();
            }

            // ---- full ring-CNOT layer as one permutation ----
            float tmp[32];
            #pragma unroll
            for (int k = 0; k < 32; ++k) {
                const int e   = tid + k * 256;
                const int s   = e >> 12;
                const int col = (e >> 8) & 15;
                const int row = e & 255;
                tmp[k] = st[s][col][row];
            }
            __syncthreads();
            #pragma unroll
            for (int k = 0; k < 32; ++k) {
                const int e   = tid + k * 256;
                const int s   = e >> 12;
                const int col = (e >> 8) & 15;
                const int row = e & 255;
                st[s][col][perm[row]] = tmp[k];   // new[sigma(i)] = old[i]
            }
            __syncthreads();
        }

        // accumulate Frobenius inner product for this chunk of columns
        for (int e = tid; e < 16 * 256; e += 256) {
            const int col = e >> 8;
            const int row = e & 255;
            acc += st[0][col][row] * st[1][col][row];
        }
        __syncthreads();
    }

    // workgroup reduction
    red[tid] = acc;
    __syncthreads();
    #pragma unroll
    for (int off = 128; off > 0; off >>= 1) {
        if (tid < off) red[tid] += red[tid + off];
        __syncthreads();
    }
    if (tid == 0) out[b] = red[0] * (1.0f / 256.0f);
}

// ---------------------------------------------------------------------------
extern "C" void kernel_launch(void* const* d_in, const int* in_sizes, int n_in,
                              void* d_out, int out_size, void* d_ws, size_t ws_size,
                              hipStream_t stream)
{
    const float* x1 = (const float*)d_in[0];
    const float* x2 = (const float*)d_in[1];
    const float* W1 = (const float*)d_in[2];
    const float* b1 = (const float*)d_in[3];
    const float* W2 = (const float*)d_in[4];
    const float* b2 = (const float*)d_in[5];
    const float* W3 = (const float*)d_in[6];
    const float* b3 = (const float*)d_in[7];

    float* angles = (float*)d_ws;   // [2][128][16] f32 = 16 KB
    float* out    = (float*)d_out;  // 128 f32

    mlp_wmma_kernel<<<16, 32, 0, stream>>>(x1, x2, W1, b1, W2, b2, W3, b3, angles);
    circuit_kernel<<<128, 256, 0, stream>>>(angles, out);
}